// ParametricInterpolation_26929444946698
// MI455X (gfx1250) — compile-verified
//
#include <hip/hip_runtime.h>
#include <hip/hip_bf16.h>

typedef __attribute__((ext_vector_type(2))) float v2f;
typedef __attribute__((ext_vector_type(8))) float v8f;

#define B_DIM 16
#define SIG_LEN 2048

// ---------------------------------------------------------------------------
// Kernel 1: curve_val = (params / SCALER) @ powers   via V_WMMA_F32_16X16X4_F32
// One wave (32 threads) computes one 16(batch) x 16(col) f32 tile of curve_val.
// A: 16x4 f32 fragment = v2f  (lanes 0-15: M=lane, K={k0,k0+1}; lanes 16-31: K={k0+2,k0+3})
// B: 4x16 f32 fragment = v2f  (lanes 0-15: N=lane, K={k0,k0+1}; lanes 16-31: K={k0+2,k0+3})
// D: 16x16 f32 = v8f          (VGPR r: lanes 0-15 -> M=r, lanes 16-31 -> M=r+8)
// K = 5 (quartic + constant), padded to 8 -> two K=4 WMMA steps.
// ---------------------------------------------------------------------------
__global__ __launch_bounds__(32) void ParametricInterp_curve_wmma(
    const float* __restrict__ params,   // [16, 5]
    float* __restrict__ curve)          // [16, 2048]
{
    const int tile = blockIdx.x;        // 0..127 (16 columns each)
    const int lane = threadIdx.x;       // 0..31
    const int m    = lane & 15;         // row (batch) / col within tile
    const bool hi  = lane >= 16;

    // p = params / SCALER ; SCALER = {1e11, 1e7, 1e3, 1, 1}
    const float inv_scaler[5] = {1e-11f, 1e-7f, 1e-3f, 1.0f, 1.0f};

    // ---- A fragments (coefficients) ----
    // step 0 covers K = 0..3  (i^4, i^3, i^2, i^1 coefficients)
    const int k0 = hi ? 2 : 0;
    v2f a_s0, a_s1;
    a_s0.x = params[m * 5 + k0 + 0] * inv_scaler[k0 + 0];
    a_s0.y = params[m * 5 + k0 + 1] * inv_scaler[k0 + 1];
    // step 1 covers K = 4..7; only K=4 (constant coeff) is non-zero
    a_s1.x = hi ? 0.0f : params[m * 5 + 4] * inv_scaler[4];
    a_s1.y = 0.0f;

    // ---- B fragments (power basis for this column) ----
    const int   col = tile * 16 + m;
    const float fi  = (float)col;
    const float fi2 = fi * fi;
    const float fi3 = fi2 * fi;
    const float fi4 = fi2 * fi2;
    v2f b_s0, b_s1;
    b_s0.x = hi ? fi2 : fi4;            // K = k0     -> i^(4-k0)
    b_s0.y = hi ? fi  : fi3;            // K = k0 + 1 -> i^(3-k0)
    b_s1.x = hi ? 0.0f : 1.0f;          // K = 4 -> i^0 ; K=6 -> 0
    b_s1.y = 0.0f;                      // K = 5 / 7 -> 0

    v8f c = {};
    // 8 args: (neg_a, A, neg_b, B, c_mod, C, reuse_a, reuse_b)
    c = __builtin_amdgcn_wmma_f32_16x16x4_f32(false, a_s0, false, b_s0,
                                              (short)0, c, false, false);
    c = __builtin_amdgcn_wmma_f32_16x16x4_f32(false, a_s1, false, b_s1,
                                              (short)0, c, false, false);

    // ---- scatter D: VGPR r holds row M = r (+8 for hi lanes), col = tile*16 + (lane&15)
    const int mbase = hi ? 8 : 0;
#pragma unroll
    for (int r = 0; r < 8; ++r) {
        curve[(mbase + r) * SIG_LEN + col] = c[r];
    }
}

// ---------------------------------------------------------------------------
// Kernel 2: windowed soft-gather + interpolation.
// In f32, exp(-diff^4) == 0 for |diff| >= 4 (exp(-256) underflows), and diff is
// integer-valued because raw_pos = i - round(curve_val).  k1 support: |diff|<=3,
// k2 support: |diff-1|<=3.  A 10-tap window [c-5, c+4] covers both supports
// with margin; out-of-range j contribute nothing, exactly as in the dense sum.
// ---------------------------------------------------------------------------
__global__ __launch_bounds__(256) void ParametricInterp_gather(
    const float* __restrict__ x,        // [16, 2048]
    const float* __restrict__ curve,    // [16, 2048]
    float* __restrict__ out)            // [16, 2048]
{
    const int gid = blockIdx.x * blockDim.x + threadIdx.x;
    if (gid >= B_DIM * SIG_LEN) return;
    const int b = gid >> 11;            // / 2048
    const int i = gid & (SIG_LEN - 1);

    const float cv = curve[gid];
    const float ci = rintf(cv);         // round-half-even, matches jnp.round
    const float d  = cv - ci;

    float np = (float)i - ci;           // integer-valued
    np = fminf(fmaxf(np, 1.0f), 2047.0f);
    const int c = (int)np;

    const float* xb = x + b * SIG_LEN;
    float n1 = 0.0f, w1 = 0.0f, n2 = 0.0f, w2 = 0.0f;
#pragma unroll
    for (int t = -5; t <= 4; ++t) {
        const int j = c + t;
        if (j < 0 || j > SIG_LEN - 1) continue;
        const float diff = np - (float)j;
        float q1 = diff * diff;  q1 = q1 * q1;          // diff^4
        const float e  = diff - 1.0f;
        float q2 = e * e;        q2 = q2 * q2;          // (diff-1)^4
        const float k1 = __expf(-q1);                   // v_exp_f32
        const float k2 = __expf(-q2);
        const float xv = xb[j];
        n1 = fmaf(xv, k1, n1);  w1 += k1;
        n2 = fmaf(xv, k2, n2);  w2 += k2;
    }
    // w1 >= exp(0)=1 (j=c in range since np in [1,2047]); w2 >= 1 (j=c-1 >= 0).
    const float a1 = n1 / w1;
    const float a2 = n2 / w2;
    out[gid] = a1 * (1.0f - d) + a2 * d;
}

extern "C" void kernel_launch(void* const* d_in, const int* in_sizes, int n_in,
                              void* d_out, int out_size, void* d_ws, size_t ws_size,
                              hipStream_t stream) {
    const float* x      = (const float*)d_in[0];   // [16, 2048]
    const float* params = (const float*)d_in[1];   // [16, 5]
    float* out   = (float*)d_out;                  // first output:  out       [16,2048]
    float* curve = (float*)d_out + B_DIM * SIG_LEN; // second output: curve_val [16,2048]

    // Phase 1: WMMA f32 polynomial evaluation -> curve_val (second half of d_out)
    ParametricInterp_curve_wmma<<<SIG_LEN / 16, 32, 0, stream>>>(params, curve);
    // Phase 2: 10-tap windowed soft-gather + lerp -> out (first half of d_out)
    ParametricInterp_gather<<<(B_DIM * SIG_LEN) / 256, 256, 0, stream>>>(x, curve, out);
}